// Linear_act_sp_56014963474597
// MI455X (gfx1250) — compile-verified
//
#include <hip/hip_runtime.h>

// CDNA5 / gfx1250, wave32. 2:4 sparse bf16 GEMM via V_SWMMAC_F32_16X16X64_BF16.
// Pipeline: colmax -> scales -> pack(x/s, 2:4, bf16 + 2-bit indices)
//           -> convert(w*s -> bf16) -> SWMMAC GEMM (64x64 wave tile).

typedef __attribute__((ext_vector_type(8)))  __bf16 v8bf;
typedef __attribute__((ext_vector_type(16))) __bf16 v16bf;
typedef __attribute__((ext_vector_type(32))) __bf16 v32bf;
typedef __attribute__((ext_vector_type(8)))  float  v8f;

constexpr int D_IN  = 4096;
constexpr int D_OUT = 4096;
constexpr int KP    = D_IN / 2;    // packed K
constexpr int KC    = D_IN / 32;   // index words per row (8 groups per word)

// ---------------- pass 0: zero the column-max accumulators ----------------
__global__ void init_max_kernel(unsigned int* __restrict__ p, int n) {
    int i = blockIdx.x * blockDim.x + threadIdx.x;
    if (i < n) p[i] = 0u;
}

// ---------------- pass 1: per-column max(|.|) ----------------
__global__ void colmax_kernel(const float* __restrict__ src,
                              unsigned int* __restrict__ dst,
                              int ncols, int rowsPerChunk, int nrows) {
    int col = blockIdx.x * blockDim.x + threadIdx.x;
    if (col >= ncols) return;
    int r0 = blockIdx.y * rowsPerChunk;
    int r1 = min(r0 + rowsPerChunk, nrows);
    float m = 0.f;
    for (int r = r0; r < r1; ++r)
        m = fmaxf(m, fabsf(src[(size_t)r * ncols + col]));
    atomicMax(&dst[col], __float_as_uint(m));
}

// ---------------- pass 2: s = sqrt(max_act / clip(max_w, 1e-8)) ----------------
__global__ void scale_kernel(const unsigned int* __restrict__ maxact,
                             const unsigned int* __restrict__ maxw,
                             float* __restrict__ s, float* __restrict__ invs, int n) {
    int i = blockIdx.x * blockDim.x + threadIdx.x;
    if (i >= n) return;
    float ma = __uint_as_float(maxact[i]);
    float mw = fmaxf(__uint_as_float(maxw[i]), 1e-8f);
    float sv = sqrtf(ma / mw);
    s[i]    = sv;
    invs[i] = 1.0f / sv;
}

// ---------------- pass 3: pack x/s with 2:4 sparsity ----------------
// One thread per (row, 32-K chunk): emits 16 packed bf16 values (kept pair per
// group of 4, in index order so idx0 < idx1) and one u32 holding 8 nibbles of
// 2-bit index pairs, nibble g at bits [4g+3:4g] per ISA 7.12.4.
__global__ __launch_bounds__(256) void pack_a_kernel(
    const float* __restrict__ X, const float* __restrict__ invs,
    __bf16* __restrict__ Apk, unsigned int* __restrict__ Aidx, int M) {
    int t = blockIdx.x * blockDim.x + threadIdx.x;
    if (t >= M * KC) return;
    int m = t >> 7;        // / KC (KC == 128)
    int c = t & (KC - 1);

    const float* px  = X + (size_t)m * D_IN + 32 * c;
    const float* pis = invs + 32 * c;

    float g[32];
#pragma unroll
    for (int e = 0; e < 32; ++e) g[e] = px[e] * pis[e];

    __bf16 outv[16];
    unsigned code = 0;
#pragma unroll
    for (int gi = 0; gi < 8; ++gi) {
        const float* q = g + 4 * gi;
        float a[4] = {fabsf(q[0]), fabsf(q[1]), fabsf(q[2]), fabsf(q[3])};
        int i0 = -1, i1 = -1;
#pragma unroll
        for (int i = 0; i < 4; ++i) {
            int rank = 0;
#pragma unroll
            for (int j = 0; j < 4; ++j) {
                if (j == i) continue;
                rank += (a[j] > a[i]) || ((a[j] == a[i]) && (j < i));
            }
            if (rank < 2) { if (i0 < 0) i0 = i; else i1 = i; }
        }
        outv[2 * gi]     = (__bf16)q[i0];
        outv[2 * gi + 1] = (__bf16)q[i1];
        code |= (unsigned)(i0 | (i1 << 2)) << (4 * gi);
    }

    __bf16* pd = Apk + (size_t)m * KP + 16 * c;
    v8bf lo, hi;
#pragma unroll
    for (int e = 0; e < 8; ++e) { lo[e] = outv[e]; hi[e] = outv[8 + e]; }
    *(v8bf*)(pd)     = lo;
    *(v8bf*)(pd + 8) = hi;
    Aidx[(size_t)m * KC + c] = code;
}

// ---------------- pass 4: Bw = bf16(w * s) ----------------
__global__ __launch_bounds__(256) void pack_b_kernel(
    const float* __restrict__ W, const float* __restrict__ svec,
    __bf16* __restrict__ Bw) {
    int t = blockIdx.x * blockDim.x + threadIdx.x;   // one thread per 8 elems
    if (t >= D_OUT * (D_IN / 8)) return;
    int n  = t >> 9;               // / (D_IN/8)
    int kc = (t & 511) * 8;
    const float* p = W + (size_t)n * D_IN + kc;
    v8bf o;
#pragma unroll
    for (int e = 0; e < 8; ++e) o[e] = (__bf16)(p[e] * svec[kc + e]);
    *(v8bf*)(Bw + (size_t)n * D_IN + kc) = o;
}

// ---------------- pass 5: SWMMAC GEMM ----------------
// Block = 256 threads = 8 waves (2x4 grid), output tile 128(M) x 256(N).
// Each wave owns a square 64x64 region = 4x4 subtiles of 16x16 for maximal
// A/B fragment reuse (16 swmmac per 4 A-frag + 4 B-frag loads), K step 64.
//
// Packed A frag (16x32 bf16): lane = 16*khalf + m; elems 0..7 = packed K
// kp+8*khalf.., elems 8..15 = packed K kp+16+8*khalf..
// Index VGPR: lane 16*h + m holds 8 nibbles for expanded-K half h -> word
// Aidx[row][k0/32 + khalf].
// B frag (64x16 bf16): lane = 16*h + n; elems 0..15 = K k0+16h.., elems
// 16..31 = K k0+32+16h.. (contiguous runs of 16 in row-major Bw).
__global__ __launch_bounds__(256) void gemm_swmmac_kernel(
    const __bf16* __restrict__ Apk, const unsigned int* __restrict__ Aidx,
    const __bf16* __restrict__ Bw, float* __restrict__ out, int M) {
    const int tid   = threadIdx.x;
    const int lane  = tid & 31;
    const int wave  = tid >> 5;
    const int wy    = wave & 1;     // M offset 64*wy
    const int wx    = wave >> 1;    // N offset 64*wx
    const int lrow  = lane & 15;
    const int khalf = lane >> 4;

    const int mTile = blockIdx.y * 128;
    const int nTile = blockIdx.x * 256;

    v8f acc[4][4];
#pragma unroll
    for (int i = 0; i < 4; ++i)
#pragma unroll
        for (int j = 0; j < 4; ++j) acc[i][j] = 0.0f;

    const __bf16* aRow[4];
    const unsigned int* aIdxRow[4];
#pragma unroll
    for (int i = 0; i < 4; ++i) {
        int r = mTile + 64 * wy + 16 * i + lrow;
        aRow[i]    = Apk + (size_t)r * KP;
        aIdxRow[i] = Aidx + (size_t)r * KC;
    }
    const __bf16* bRow[4];
#pragma unroll
    for (int j = 0; j < 4; ++j)
        bRow[j] = Bw + (size_t)(nTile + 64 * wx + 16 * j + lrow) * D_IN;

    for (int k0 = 0; k0 < D_IN; k0 += 64) {
        const int kp = k0 >> 1;

        // pull next K-step's operand lines toward GL2 (global_prefetch_b8)
        if (k0 + 64 < D_IN) {
#pragma unroll
            for (int i = 0; i < 4; ++i)
                __builtin_prefetch(aRow[i] + kp + 32, 0, 3);
#pragma unroll
            for (int j = 0; j < 4; ++j)
                __builtin_prefetch(bRow[j] + k0 + 64, 0, 3);
        }

        v16bf afrag[4];
        int   idx[4];
#pragma unroll
        for (int i = 0; i < 4; ++i) {
            v8bf lo = *(const v8bf*)(aRow[i] + kp + 8 * khalf);
            v8bf hi = *(const v8bf*)(aRow[i] + kp + 16 + 8 * khalf);
#pragma unroll
            for (int e = 0; e < 8; ++e) {
                afrag[i][e]     = lo[e];
                afrag[i][8 + e] = hi[e];
            }
            idx[i] = (int)aIdxRow[i][(k0 >> 5) + khalf];
        }

#pragma unroll
        for (int j = 0; j < 4; ++j) {
            v32bf bfrag;
            v16bf lo = *(const v16bf*)(bRow[j] + k0 + 16 * khalf);
            v16bf hi = *(const v16bf*)(bRow[j] + k0 + 32 + 16 * khalf);
#pragma unroll
            for (int e = 0; e < 16; ++e) {
                bfrag[e]      = lo[e];
                bfrag[16 + e] = hi[e];
            }
#pragma unroll
            for (int i = 0; i < 4; ++i)
                acc[i][j] = __builtin_amdgcn_swmmac_f32_16x16x64_bf16(
                    /*neg_a=*/false, afrag[i],
                    /*neg_b=*/false, bfrag,
                    acc[i][j], idx[i],
                    /*reuse_a=*/false, /*reuse_b=*/false);
        }
    }

    // C/D layout: elem r at lane l is C[8*(l>>4)+r][l&15]
#pragma unroll
    for (int i = 0; i < 4; ++i) {
        const int rowBase = mTile + 64 * wy + 16 * i + 8 * khalf;
#pragma unroll
        for (int j = 0; j < 4; ++j) {
            const int col = nTile + 64 * wx + 16 * j + lrow;
#pragma unroll
            for (int r = 0; r < 8; ++r)
                out[(size_t)(rowBase + r) * D_OUT + col] = acc[i][j][r];
        }
    }
}

extern "C" void kernel_launch(void* const* d_in, const int* in_sizes, int n_in,
                              void* d_out, int out_size, void* d_ws, size_t ws_size,
                              hipStream_t stream) {
    (void)n_in; (void)out_size; (void)ws_size;

    const float* x = (const float*)d_in[0];   // [M, D_IN] fp32
    const float* w = (const float*)d_in[1];   // [D_OUT, D_IN] fp32
    float* out = (float*)d_out;               // [M, D_OUT] fp32

    const int M = in_sizes[0] / D_IN;         // 16384

    // Workspace layout (≈104 MB):
    //   [0)      maxact : 4096 u32
    //   [16KB)   maxw   : 4096 u32
    //   [32KB)   s      : 4096 f32
    //   [48KB)   invs   : 4096 f32
    //   [64KB)   Aidx   : M*KC u32            (8 MB)
    //   [+8MB)   Bw     : D_OUT*D_IN bf16     (32 MB)
    //   [+32MB)  Apk    : M*KP bf16           (64 MB)
    char* ws = (char*)d_ws;
    unsigned int* maxact = (unsigned int*)(ws);
    unsigned int* maxw   = (unsigned int*)(ws + (16 << 10));
    float*        s      = (float*)(ws + (32 << 10));
    float*        invs   = (float*)(ws + (48 << 10));
    unsigned int* Aidx   = (unsigned int*)(ws + (64 << 10));
    __bf16*       Bw     = (__bf16*)(ws + (64 << 10) + (size_t)M * KC * 4);
    __bf16*       Apk    = (__bf16*)((char*)Bw + (size_t)D_OUT * D_IN * 2);

    init_max_kernel<<<(2 * D_IN + 255) / 256, 256, 0, stream>>>(maxact, 2 * D_IN);
    colmax_kernel<<<dim3(D_IN / 256, (M + 255) / 256), 256, 0, stream>>>(
        x, maxact, D_IN, 256, M);
    colmax_kernel<<<dim3(D_IN / 256, (D_OUT + 255) / 256), 256, 0, stream>>>(
        w, maxw, D_IN, 256, D_OUT);
    scale_kernel<<<D_IN / 256, 256, 0, stream>>>(maxact, maxw, s, invs, D_IN);

    pack_a_kernel<<<(M * KC + 255) / 256, 256, 0, stream>>>(x, invs, Apk, Aidx, M);
    pack_b_kernel<<<(D_OUT * (D_IN / 8) + 255) / 256, 256, 0, stream>>>(w, s, Bw);

    gemm_swmmac_kernel<<<dim3(D_OUT / 256, M / 128), 256, 0, stream>>>(
        Apk, Aidx, Bw, out, M);
}